// SGE_42391327212246
// MI455X (gfx1250) — compile-verified
//
#include <hip/hip_runtime.h>
#include <hip/hip_bf16.h>

typedef __attribute__((ext_vector_type(2))) float v2f;
typedef __attribute__((ext_vector_type(8))) float v8f;

#define EPS      0.1f
#define INV_EPS  10.0f
#define THRESH   0.1f
#define NP       32
#define SD       16
#define NNODES   10000
#define FEAT     256
#define EMB      512
#define NEDGES   4096
#define LOG_MU  -3.4657356f   /* log(1/32 + 1e-8) */

// ---------------------------------------------------------------------------
// Encoder GEMM: Z[10000,512] = X[10000,256] @ W[256,512], exact fp32 via
// V_WMMA_F32_16X16X4_F32. One wave per 16x16 output tile; 64 K-steps.
// A frag (16x4): lane L holds A[M=L&15][K = 2*(L>>4) + j] in vgpr j (float2).
// B frag (4x16): lane L holds B[K = 2*(L>>4) + r][N=L&15] in vgpr r.
// C/D (16x16):   vgpr r -> M = r + 8*(L>>4), N = L&15.
// ---------------------------------------------------------------------------
__global__ __launch_bounds__(256)
void sge_gemm_wmma(const float* __restrict__ X, const float* __restrict__ W,
                   float* __restrict__ Z)
{
  const int wave = threadIdx.x >> 5;
  const int lane = threadIdx.x & 31;
  const int tile = blockIdx.x * 8 + wave;      // 0..19999 (625 x 32 tiles)
  const int tM = tile >> 5;                    // 0..624
  const int tN = tile & 31;                    // 0..31
  const int h   = lane >> 4;
  const int l15 = lane & 15;

  const float* arow = X + (tM * 16 + l15) * FEAT + 2 * h;
  const float* bcol = W + (2 * h) * EMB + tN * 16 + l15;

  v8f acc = {};
#pragma unroll 4
  for (int ks = 0; ks < FEAT / 4; ++ks) {
    v2f af = *(const v2f*)(arow + ks * 4);
    v2f bf;
    bf.x = bcol[(ks * 4 + 0) * EMB];
    bf.y = bcol[(ks * 4 + 1) * EMB];
    acc = __builtin_amdgcn_wmma_f32_16x16x4_f32(false, af, false, bf,
                                                (short)0, acc, false, false);
  }
  float* zrow = Z + (tM * 16 + 8 * h) * EMB + tN * 16 + l15;
#pragma unroll
  for (int r = 0; r < 8; ++r) zrow[r * EMB] = acc[r];
}

// ---------------------------------------------------------------------------
// C[i][j] = ||a_i||^2 + ||b_j||^2 - 2 a_i.b_j  for 32 points in R^16.
// Dot part = (32x16)@(16x32): 2x2 tiles of 16x16, K=16 -> 4 wmma steps each.
// Stored to LDS with row stride 33 (bank-conflict free rows AND columns).
// Must be entered by a full wave (EXEC all ones for WMMA).
// ---------------------------------------------------------------------------
__device__ __forceinline__
void compute_cost_matrix(const float* __restrict__ a, const float* __restrict__ b,
                         float* sC, float* sNa, float* sNb, int lane)
{
  const int h = lane >> 4, l15 = lane & 15;
  float na = 0.f, nb = 0.f;
  const float* ar = a + lane * SD;
  const float* br = b + lane * SD;
#pragma unroll
  for (int d = 0; d < SD; ++d) { na += ar[d] * ar[d]; nb += br[d] * br[d]; }
  sNa[lane] = na; sNb[lane] = nb;
  __syncthreads();

#pragma unroll
  for (int ti = 0; ti < 2; ++ti) {
#pragma unroll
    for (int tj = 0; tj < 2; ++tj) {
      v8f acc = {};
#pragma unroll
      for (int ks = 0; ks < 4; ++ks) {
        v2f af = *(const v2f*)(a + (ti * 16 + l15) * SD + ks * 4 + 2 * h);
        v2f bf = *(const v2f*)(b + (tj * 16 + l15) * SD + ks * 4 + 2 * h);
        acc = __builtin_amdgcn_wmma_f32_16x16x4_f32(false, af, false, bf,
                                                    (short)0, acc, false, false);
      }
#pragma unroll
      for (int r = 0; r < 8; ++r) {
        int i = ti * 16 + r + 8 * h;
        int j = tj * 16 + l15;
        sC[i * 33 + j] = sNa[i] + sNb[j] - 2.0f * acc[r];
      }
    }
  }
  __syncthreads();
}

__device__ __forceinline__
void edge_nodes(int e, const int* ep, const int* en, int& nx, int& ny)
{
  if (e < NEDGES) { nx = ep[e]; ny = ep[NEDGES + e]; }
  else            { int q = e - NEDGES; nx = en[q]; ny = en[NEDGES + q]; }
}

// ---------------------------------------------------------------------------
// One Sinkhorn iteration for all 8192 edges. One wave per edge.
// Skips (uniformly) if this batch's global done flag is set.
// ---------------------------------------------------------------------------
__global__ __launch_bounds__(32)
void sinkhorn_iter(const float* __restrict__ Z,
                   const int* __restrict__ ep, const int* __restrict__ en,
                   float* __restrict__ U, float* __restrict__ V,
                   float* __restrict__ errP, float* __restrict__ errN,
                   const int* __restrict__ doneP, const int* __restrict__ doneN)
{
  __shared__ float sC[32 * 33];
  __shared__ float sNa[32], sNb[32], sU[32], sV[32], sUn[32], sRed[32];

  const int e = blockIdx.x;
  const int lane = threadIdx.x;
  const int* done = (e < NEDGES) ? doneP : doneN;
  if (*done) return;                        // wave-uniform

  int nx, ny; edge_nodes(e, ep, en, nx, ny);
  compute_cost_matrix(Z + (size_t)nx * EMB, Z + (size_t)ny * EMB,
                      sC, sNa, sNb, lane);

  float u = U[e * 32 + lane];
  float v = V[e * 32 + lane];
  sU[lane] = u; sV[lane] = v;
  __syncthreads();

  // u update: lane owns row i; logsumexp over j of (-C+u_i+v_j)/eps
  float mx = -3.4e38f;
#pragma unroll
  for (int j = 0; j < 32; ++j)
    mx = fmaxf(mx, (sV[j] + u - sC[lane * 33 + j]) * INV_EPS);
  float s = 0.f;
#pragma unroll
  for (int j = 0; j < 32; ++j)
    s += expf((sV[j] + u - sC[lane * 33 + j]) * INV_EPS - mx);
  float lse = mx + logf(s);
  float un = EPS * (LOG_MU - lse) + u;
  sUn[lane] = un;
  sRed[lane] = fabsf(un - u);
  __syncthreads();

  // v update: lane owns column j; uses u_new (reference M(u_new, v))
  mx = -3.4e38f;
#pragma unroll
  for (int i = 0; i < 32; ++i)
    mx = fmaxf(mx, (sUn[i] + v - sC[i * 33 + lane]) * INV_EPS);
  s = 0.f;
#pragma unroll
  for (int i = 0; i < 32; ++i)
    s += expf((sUn[i] + v - sC[i * 33 + lane]) * INV_EPS - mx);
  lse = mx + logf(s);
  float vn = EPS * (LOG_MU - lse) + v;

  U[e * 32 + lane] = un;
  V[e * 32 + lane] = vn;

  // err_e = sum_i |u_new - u|  (fixed-order LDS tree -> deterministic)
  __syncthreads();
  for (int off = 16; off > 0; off >>= 1) {
    if (lane < off) sRed[lane] += sRed[lane + off];
    __syncthreads();
  }
  if (lane == 0) {
    if (e < NEDGES) errP[e] = sRed[0];
    else            errN[e - NEDGES] = sRed[0];
  }
}

// Global (per-batch) early-exit decision: done |= mean(err) < THRESH.
__global__ __launch_bounds__(256)
void err_update(const float* __restrict__ errP, const float* __restrict__ errN,
                int* doneP, int* doneN)
{
  __shared__ float sp[256], sn[256];
  const int t = threadIdx.x;
  float ap = 0.f, an = 0.f;
  for (int i = t; i < NEDGES; i += 256) { ap += errP[i]; an += errN[i]; }
  sp[t] = ap; sn[t] = an;
  __syncthreads();
  for (int off = 128; off > 0; off >>= 1) {
    if (t < off) { sp[t] += sp[t + off]; sn[t] += sn[t + off]; }
    __syncthreads();
  }
  if (t == 0) {
    if (sp[0] * (1.0f / NEDGES) < THRESH) *doneP = 1;
    if (sn[0] * (1.0f / NEDGES) < THRESH) *doneN = 1;
  }
}

// pi = exp(M(u,v)); cost = sum(pi*C); term = pos: (-cost)^2, neg: exp(-cost)
__global__ __launch_bounds__(32)
void final_energy(const float* __restrict__ Z,
                  const int* __restrict__ ep, const int* __restrict__ en,
                  const float* __restrict__ U, const float* __restrict__ V,
                  float* __restrict__ terms)
{
  __shared__ float sC[32 * 33];
  __shared__ float sNa[32], sNb[32], sV[32], sRed[32];
  const int e = blockIdx.x, lane = threadIdx.x;

  int nx, ny; edge_nodes(e, ep, en, nx, ny);
  compute_cost_matrix(Z + (size_t)nx * EMB, Z + (size_t)ny * EMB,
                      sC, sNa, sNb, lane);

  float u = U[e * 32 + lane];
  sV[lane] = V[e * 32 + lane];
  __syncthreads();

  float part = 0.f;
#pragma unroll
  for (int j = 0; j < 32; ++j) {
    float c = sC[lane * 33 + j];
    part += expf((sV[j] + u - c) * INV_EPS) * c;
  }
  sRed[lane] = part;
  __syncthreads();
  for (int off = 16; off > 0; off >>= 1) {
    if (lane < off) sRed[lane] += sRed[lane + off];
    __syncthreads();
  }
  if (lane == 0) {
    float energy = -sRed[0];
    terms[e] = (e < NEDGES) ? energy * energy : expf(energy);
  }
}

__global__ __launch_bounds__(256)
void loss_reduce(const float* __restrict__ terms, float* __restrict__ out)
{
  __shared__ float s[256];
  const int t = threadIdx.x;
  float a = 0.f;
  for (int i = t; i < 2 * NEDGES; i += 256) a += terms[i];
  s[t] = a;
  __syncthreads();
  for (int off = 128; off > 0; off >>= 1) {
    if (t < off) s[t] += s[t + off];
    __syncthreads();
  }
  if (t == 0) out[0] = s[0] * (1.0f / NEDGES);
}

__global__ __launch_bounds__(256)
void init_state(float* U, float* V, int* doneP, int* doneN)
{
  const int i = blockIdx.x * blockDim.x + threadIdx.x;
  if (i < 2 * NEDGES * 32) { U[i] = 0.f; V[i] = 0.f; }
  if (i == 0) { *doneP = 0; *doneN = 0; }
}

// ---------------------------------------------------------------------------
extern "C" void kernel_launch(void* const* d_in, const int* in_sizes, int n_in,
                              void* d_out, int out_size, void* d_ws, size_t ws_size,
                              hipStream_t stream)
{
  (void)in_sizes; (void)n_in; (void)out_size; (void)ws_size;
  const float* x  = (const float*)d_in[0];
  const float* W  = (const float*)d_in[1];
  const int*   ep = (const int*)d_in[2];
  const int*   en = (const int*)d_in[3];
  float* out = (float*)d_out;

  char* ws = (char*)d_ws;
  size_t off = 0;
  auto wsalloc = [&](size_t bytes) -> void* {
    void* p = ws + off;
    off = (off + bytes + 255) & ~(size_t)255;
    return p;
  };
  float* Z     = (float*)wsalloc((size_t)NNODES * EMB * sizeof(float)); // 20.5 MB
  float* U     = (float*)wsalloc((size_t)2 * NEDGES * 32 * sizeof(float));
  float* V     = (float*)wsalloc((size_t)2 * NEDGES * 32 * sizeof(float));
  float* errP  = (float*)wsalloc((size_t)NEDGES * sizeof(float));
  float* errN  = (float*)wsalloc((size_t)NEDGES * sizeof(float));
  float* terms = (float*)wsalloc((size_t)2 * NEDGES * sizeof(float));
  int*   doneP = (int*)wsalloc(sizeof(int));
  int*   doneN = (int*)wsalloc(sizeof(int));

  init_state<<<(2 * NEDGES * 32 + 255) / 256, 256, 0, stream>>>(U, V, doneP, doneN);

  // 625*32 = 20000 tiles, 8 waves/block
  sge_gemm_wmma<<<2500, 256, 0, stream>>>(x, W, Z);

  for (int t = 0; t < 10; ++t) {
    sinkhorn_iter<<<2 * NEDGES, 32, 0, stream>>>(Z, ep, en, U, V,
                                                 errP, errN, doneP, doneN);
    err_update<<<1, 256, 0, stream>>>(errP, errN, doneP, doneN);
  }

  final_energy<<<2 * NEDGES, 32, 0, stream>>>(Z, ep, en, U, V, terms);
  loss_reduce<<<1, 256, 0, stream>>>(terms, out);
}